// MS_SPS_79216376807543
// MI455X (gfx1250) — compile-verified
//
#include <hip/hip_runtime.h>
#include <hip/hip_bf16.h>

// Problem constants (from reference setup_inputs)
#define TT 16
#define BB 512
#define CC 2
#define VV 25
#define EE 256
#define NTOT (TT * BB * VV)          // 204800 elements per channel for BN stats
#define NCOL (BB * VV)               // 12800 columns per timestep (multiple of 16)
#define NTILE (NCOL / 16)            // 800
#define ETILE (EE / 16)              // 16
#define BN_EPS 1e-5f

typedef float v2f __attribute__((ext_vector_type(2)));
typedef float v8f __attribute__((ext_vector_type(8)));

// ---------------------------------------------------------------------------
// Kernel 1: partial sums for BN statistics (deterministic two-stage reduction)
// Each of 256 blocks writes 5 partials: s0, s1, q00, q01, q11
// ---------------------------------------------------------------------------
__global__ void __launch_bounds__(256)
stats_partial_kernel(const float* __restrict__ x, float* __restrict__ partial) {
    __shared__ float sm[5][256];
    const int tid = threadIdx.x;
    float a0 = 0.f, a1 = 0.f, a00 = 0.f, a01 = 0.f, a11 = 0.f;
    for (int i = blockIdx.x * blockDim.x + tid; i < NTOT; i += gridDim.x * blockDim.x) {
        const int row = i / VV;          // (t*B + b)
        const int v   = i - row * VV;
        const float x0 = x[row * (CC * VV) + v];
        const float x1 = x[row * (CC * VV) + VV + v];
        a0 += x0; a1 += x1;
        a00 += x0 * x0; a01 += x0 * x1; a11 += x1 * x1;
    }
    sm[0][tid] = a0; sm[1][tid] = a1; sm[2][tid] = a00; sm[3][tid] = a01; sm[4][tid] = a11;
    __syncthreads();
    for (int s = 128; s > 0; s >>= 1) {
        if (tid < s) {
            #pragma unroll
            for (int k = 0; k < 5; k++) sm[k][tid] += sm[k][tid + s];
        }
        __syncthreads();
    }
    if (tid < 5) partial[blockIdx.x * 5 + tid] = sm[tid][0];
}

// ---------------------------------------------------------------------------
// Kernel 2: finalize stats -> per-channel fused BN scale/bias
// scale_e = gamma_e * rsqrt(var_e + eps); bias_e = beta_e - mean_e * scale_e
// ---------------------------------------------------------------------------
__global__ void __launch_bounds__(256)
stats_finalize_kernel(const float* __restrict__ partial,
                      const float* __restrict__ W,
                      const float* __restrict__ gamma,
                      const float* __restrict__ beta,
                      float* __restrict__ scale,
                      float* __restrict__ bias) {
    __shared__ float tot[5];
    const int tid = threadIdx.x;
    if (tid < 5) {
        float s = 0.f;
        for (int i = 0; i < 256; i++) s += partial[i * 5 + tid];
        tot[tid] = s;
    }
    __syncthreads();
    const float invN = 1.0f / (float)NTOT;
    const float w0 = W[2 * tid], w1 = W[2 * tid + 1];
    const float mean = (w0 * tot[0] + w1 * tot[1]) * invN;
    const float ey2  = (w0 * w0 * tot[2] + 2.0f * w0 * w1 * tot[3] + w1 * w1 * tot[4]) * invN;
    const float var  = ey2 - mean * mean;
    const float sc   = gamma[tid] * rsqrtf(var + BN_EPS);
    scale[tid] = sc;
    bias[tid]  = beta[tid] - mean * sc;
}

// ---------------------------------------------------------------------------
// Kernel 3: fused conv1x1 (WMMA f32 16x16x4, K padded 2->4) + BN + LIF scan.
// One wave per (E-tile, N-tile); LIF membrane state v lives in 8 VGPRs
// (matching WMMA C/D layout) across all 16 timesteps. Output-store bound:
// 210 MB @ 23.3 TB/s ~ 9 us floor.
// ---------------------------------------------------------------------------
__global__ void __launch_bounds__(256)
lif_wmma_kernel(const float* __restrict__ x,
                const float* __restrict__ W,
                const float* __restrict__ scale,
                const float* __restrict__ bias,
                float* __restrict__ out) {
    const int lane  = threadIdx.x & 31;
    const int wave  = threadIdx.x >> 5;
    const int tile  = blockIdx.x * 8 + wave;     // 12800 tiles total
    const int etile = tile & (ETILE - 1);        // tile % 16
    const int ntile = tile >> 4;                 // tile / 16
    const int lo    = lane & 15;
    const int hi    = lane >> 4;                 // 0 or 1

    const int n  = ntile * 16 + lo;              // column: (b, v) pair
    const int b  = n / VV;
    const int v  = n - b * VV;
    const int e0 = etile * 16;

    // A-matrix (16x4 f32): lanes 0-15 hold (K0,K1) = (W[e,0], W[e,1]); lanes 16-31 hold K2,K3 = 0
    const int eA = e0 + lo;
    v2f amat;
    amat.x = (hi == 0) ? W[2 * eA]     : 0.0f;
    amat.y = (hi == 0) ? W[2 * eA + 1] : 0.0f;

    // Per-lane BN scale/bias for the 8 E-rows this lane owns (t-invariant)
    float sc[8], bi[8];
    #pragma unroll
    for (int r = 0; r < 8; r++) {
        const int e = e0 + 8 * hi + r;
        sc[r] = scale[e];
        bi[r] = bias[e];
    }

    // LIF membrane state, one per owned (M,N) element
    v8f vmem = {0.f, 0.f, 0.f, 0.f, 0.f, 0.f, 0.f, 0.f};

    const int xTstride = BB * CC * VV;           // 25600 floats per timestep
    const int oTstride = BB * EE * VV;           // 3276800 floats per timestep
    const float* xp = x + b * (CC * VV) + v;     // c=0; c=1 at +VV
    float* op = out + b * (EE * VV) + (e0 + 8 * hi) * VV + v;

    #pragma unroll 1
    for (int t = 0; t < TT; t++) {
        // B-matrix (4x16 f32): lanes 0-15 hold (K0,K1) = (x_c0, x_c1); lanes 16-31 zeros
        const float x0 = xp[0];
        const float x1 = xp[VV];
        v2f bmat;
        bmat.x = (hi == 0) ? x0 : 0.0f;
        bmat.y = (hi == 0) ? x1 : 0.0f;

        v8f czero = {0.f, 0.f, 0.f, 0.f, 0.f, 0.f, 0.f, 0.f};
        // D = A(16x4) x B(4x16) + 0  -> y tile [16e x 16n], exact f32
        v8f y = __builtin_amdgcn_wmma_f32_16x16x4_f32(
            /*neg_a=*/false, amat, /*neg_b=*/false, bmat,
            /*c_mod=*/(short)0, czero, /*reuse_a=*/false, /*reuse_b=*/false);

        #pragma unroll
        for (int r = 0; r < 8; r++) {
            const float yn = y[r] * sc[r] + bi[r];       // BN
            const float vv2 = 0.5f * (vmem[r] + yn);     // v += (y - v)/tau, tau=2
            const bool fire = (vv2 >= 1.0f);
            op[r * VV] = fire ? 1.0f : 0.0f;             // spike
            vmem[r] = fire ? 0.0f : vv2;                 // hard reset
        }
        xp += xTstride;
        op += oTstride;
    }
}

// ---------------------------------------------------------------------------
extern "C" void kernel_launch(void* const* d_in, const int* in_sizes, int n_in,
                              void* d_out, int out_size, void* d_ws, size_t ws_size,
                              hipStream_t stream) {
    const float* x     = (const float*)d_in[0];   // [16,512,2,25]
    const float* W     = (const float*)d_in[1];   // [256,2]
    const float* gamma = (const float*)d_in[2];   // [256]
    const float* beta  = (const float*)d_in[3];   // [256]
    float* out = (float*)d_out;                   // [16,512,256,25]

    float* ws      = (float*)d_ws;
    float* partial = ws;            // 256 blocks * 5 = 1280 floats
    float* scale   = ws + 1280;     // 256 floats
    float* bias    = ws + 1536;     // 256 floats

    stats_partial_kernel<<<256, 256, 0, stream>>>(x, partial);
    stats_finalize_kernel<<<1, 256, 0, stream>>>(partial, W, gamma, beta, scale, bias);

    // 12800 waves (16 E-tiles x 800 N-tiles), 8 waves per block
    lif_wmma_kernel<<<(ETILE * NTILE) / 8, 256, 0, stream>>>(x, W, scale, bias, out);
}